// PatchPositionEmbedding_30305289240876
// MI455X (gfx1250) — compile-verified
//
#include <hip/hip_runtime.h>
#include <hip/hip_bf16.h>

// fp32 WMMA fragment types (CDNA5 wave32):
//  A (16x4 f32)  = 2 VGPRs/lane, B (4x16 f32) = 2 VGPRs/lane, C/D (16x16 f32) = 8 VGPRs/lane
typedef float f32x2 __attribute__((ext_vector_type(2)));
typedef float f32x8 __attribute__((ext_vector_type(8)));

#define BATCH   64
#define CCH     3
#define HIMG    224
#define WIMG    224
#define PSZ     16
#define DMODEL  768
#define KDIM    768          // C*P*P
#define NPATCH  196          // 14*14
#define SEQ     197
#define ROWS    (BATCH * NPATCH)   // 12544

// Block tile: 64 (M) x 64 (N), K-chunk 32.  8 waves: 4 over M x 2 over N.
#define TM      64
#define TN      64
#define TK      32
#define A_STRIDE 36          // 64 rows * 36 floats  (16B-aligned rows, conflict-free frag reads)
#define B_STRIDE 68          // 32 rows * 68 floats

__global__ __launch_bounds__(256)
void vit_patch_embed_gemm(const float* __restrict__ x,
                          const float* __restrict__ Wm,
                          const float* __restrict__ bias,
                          const float* __restrict__ pos,
                          float* __restrict__ out)
{
    __shared__ float As[TM * A_STRIDE];   // 9216 B
    __shared__ float Bs[TK * B_STRIDE];   // 8704 B

    const int tid    = threadIdx.x;
    const int lane   = tid & 31;
    const int waveId = tid >> 5;
    const int waveM  = waveId & 3;        // 0..3  -> 16-row strip
    const int waveN  = waveId >> 2;       // 0..1  -> 32-col strip
    const int half   = lane >> 4;         // 0: K=0,1   1: K=2,3
    const int lsub   = lane & 15;
    const int kOff   = half * 2;

    const int rowBase = blockIdx.x * TM;  // 0..12480, exact tiling (196 blocks)
    const int nBase   = blockIdx.y * TN;  // 0..704   (12 blocks)

    // ---- A-tile load assignment: 64 rows x 32 cols, 8 floats/thread ----
    const int aRow     = tid >> 2;          // 0..63
    const int aColBase = (tid & 3) * 8;     // 0,8,16,24
    const int rG   = rowBase + aRow;        // global token row
    const int bImg = rG / NPATCH;
    const int nTok = rG - bImg * NPATCH;
    const int pi_  = nTok / 14;             // patch grid row
    const int pj_  = nTok - pi_ * 14;       // patch grid col

    // ---- B-tile load assignment: 32 rows x 64 cols, 8 floats/thread ----
    const int bRow     = tid >> 3;          // 0..31
    const int bColBase = (tid & 7) * 8;     // 0..56

    f32x8 c0 = {0.f,0.f,0.f,0.f,0.f,0.f,0.f,0.f};
    f32x8 c1 = {0.f,0.f,0.f,0.f,0.f,0.f,0.f,0.f};

    const int nA = waveN * 32;
    const int nB = nA + 16;

    for (int kBase = 0; kBase < KDIM; kBase += TK) {
        // ---- stage A (patchify fused): feature f = c*256 + pr*16 + pc ----
        {
            const int f0 = kBase + aColBase;      // multiple of 8, within one patch row
            const int ch = f0 >> 8;
            const int rem = f0 & 255;
            const int pr = rem >> 4;
            const int pc = rem & 15;              // 0 or 8
            const long gidx = ((long)(bImg * CCH + ch) * HIMG + pi_ * PSZ + pr) * WIMG
                              + pj_ * PSZ + pc;
            const float4 av0 = *(const float4*)(x + gidx);
            const float4 av1 = *(const float4*)(x + gidx + 4);
            *(float4*)&As[aRow * A_STRIDE + aColBase]     = av0;
            *(float4*)&As[aRow * A_STRIDE + aColBase + 4] = av1;
        }
        // ---- stage B: W[kBase+bRow][nBase + bColBase .. +7] ----
        {
            const long widx = (long)(kBase + bRow) * DMODEL + nBase + bColBase;
            const float4 bv0 = *(const float4*)(Wm + widx);
            const float4 bv1 = *(const float4*)(Wm + widx + 4);
            *(float4*)&Bs[bRow * B_STRIDE + bColBase]     = bv0;
            *(float4*)&Bs[bRow * B_STRIDE + bColBase + 4] = bv1;
            if (kBase + TK < KDIM) {   // pull next W chunk toward L2 (global_prefetch_b8)
                __builtin_prefetch(Wm + widx + (long)TK * DMODEL, 0, 0);
            }
        }
        __syncthreads();

        // ---- 8 x (16x16x4) WMMA steps over this K-chunk, 2 N-subtiles ----
        #pragma unroll
        for (int kk = 0; kk < TK; kk += 4) {
            f32x2 a, b0, b1;
            const int ar = (waveM * 16 + lsub) * A_STRIDE + kk + kOff;
            a.x = As[ar];
            a.y = As[ar + 1];
            const int br0 = (kk + kOff) * B_STRIDE;
            const int br1 = (kk + kOff + 1) * B_STRIDE;
            b0.x = Bs[br0 + nA + lsub];
            b0.y = Bs[br1 + nA + lsub];
            b1.x = Bs[br0 + nB + lsub];
            b1.y = Bs[br1 + nB + lsub];
            c0 = __builtin_amdgcn_wmma_f32_16x16x4_f32(false, a, false, b0,
                                                       (short)0, c0, false, false);
            c1 = __builtin_amdgcn_wmma_f32_16x16x4_f32(false, a, false, b1,
                                                       (short)0, c1, false, false);
        }
        __syncthreads();
    }

    // ---- epilogue: + bias + pos_table, scatter to (b, 1+n, d) ----
    const int colA = nBase + nA + lsub;
    const int colB = nBase + nB + lsub;
    const float biasA = bias[colA];
    const float biasB = bias[colB];
    #pragma unroll
    for (int v = 0; v < 8; ++v) {
        const int mRow = waveM * 16 + v + half * 8;
        const int r    = rowBase + mRow;
        const int bI   = r / NPATCH;
        const int s    = r - bI * NPATCH + 1;            // token index 1..196
        const long obase = ((long)bI * SEQ + s) * DMODEL;
        const long pbase = (long)s * DMODEL;
        out[obase + colA] = c0[v] + biasA + pos[pbase + colA];
        out[obase + colB] = c1[v] + biasB + pos[pbase + colB];
    }
}

__global__ __launch_bounds__(256)
void vit_class_token(const float* __restrict__ cls,
                     const float* __restrict__ pos,
                     float* __restrict__ out)
{
    const int idx = blockIdx.x * blockDim.x + threadIdx.x;  // 64*768 threads
    const int b = idx / DMODEL;
    const int d = idx - b * DMODEL;
    out[(long)b * SEQ * DMODEL + d] = cls[d] + pos[d];
}

extern "C" void kernel_launch(void* const* d_in, const int* in_sizes, int n_in,
                              void* d_out, int out_size, void* d_ws, size_t ws_size,
                              hipStream_t stream) {
    const float* x   = (const float*)d_in[0];   // (64,3,224,224)
    const float* Wm  = (const float*)d_in[1];   // (768,768)
    const float* b   = (const float*)d_in[2];   // (768,)
    const float* pos = (const float*)d_in[3];   // (197,768)
    const float* cls = (const float*)d_in[4];   // (1,1,768)
    float* out = (float*)d_out;                 // (64,197,768)

    dim3 grid(ROWS / TM, DMODEL / TN);          // 196 x 12, exact
    vit_patch_embed_gemm<<<grid, 256, 0, stream>>>(x, Wm, b, pos, out);
    vit_class_token<<<(BATCH * DMODEL) / 256, 256, 0, stream>>>(cls, pos, out);
}